// GridMask_36575941493177
// MI455X (gfx1250) — compile-verified
//
#include <hip/hip_runtime.h>
#include <stdint.h>

// Problem constants (match reference)
#define TS 1024          // S = 2*max(H,W)
#define HH 512
#define WW 512
#define BB 32
#define CC 3
#define THREADS 256
#define PIX_CHUNK 1024   // pixels per staged chunk (= 2 image rows)
#define CHUNKS 8         // chunks processed per block
#define BLOCKS_X 32      // BLOCKS_X * CHUNKS * PIX_CHUNK == H*W
#define CHUNK_BYTES (PIX_CHUNK * CC * 4)   // 12288 B
#define CHUNK_FLOATS (PIX_CHUNK * CC)      // 3072 floats

typedef float f4 __attribute__((ext_vector_type(4)));

__device__ __forceinline__ int reflect2048(int i) {
  int m = (i + 2048) & 2047;           // i >= -2048 always holds here
  return (m < TS) ? m : (2047 - m);
}

// Async copy of 16B from (base + voff) in global memory into LDS byte offset
// lds_off.  GLOBAL_LOAD_ASYNC_TO_LDS_B128, SADDR ("GVS") mode: vdst = LDS
// address VGPR, vaddr = 32-bit byte offset VGPR, saddr = 64-bit SGPR base.
// Tracked by ASYNCcnt (completes in order for loads).
__device__ __forceinline__ void async_lds_b128(uint32_t lds_off, uint64_t base,
                                               uint32_t voff) {
  asm volatile("global_load_async_to_lds_b128 %0, %1, %2"
               :
               : "v"(lds_off), "v"(voff), "s"(base)
               : "memory");
}

__global__ __launch_bounds__(THREADS) void gridmask_fused_kernel(
    const float* __restrict__ images, const float* __restrict__ angles,
    const int* __restrict__ gridblock, const int* __restrict__ start1,
    const int* __restrict__ start2, float* __restrict__ out) {
  __shared__ __align__(16) float rowLUT[TS];
  __shared__ __align__(16) float colLUT[TS];
  __shared__ __align__(16) float tile[2][CHUNK_FLOATS];   // 2 x 12 KB

  const int tid = threadIdx.x;
  const int b = blockIdx.y;

  // Per-batch scalars (uniform -> scalar loads)
  const int gb = gridblock[b];
  const int st1 = start1[b];
  const int st2 = start2[b];
  int length = (int)((float)gb * 0.6f + 0.5f);   // trunc, matches astype(int32)
  length = length < 1 ? 1 : (length > gb - 1 ? gb - 1 : length);
  const int ngb = (TS / gb) * gb;                // n * gb  (d // gb < n  <=>  d < ngb)
  const float ang = angles[b];
  const float cv = cosf(ang);
  const float sv = sinf(ang);

  const int chunk0 = blockIdx.x * CHUNKS;
  const long long imgBasePix = (long long)b * (HH * WW);
  const float* imgBase = images + imgBasePix * CC;
  float* outBase = out + imgBasePix * CC;

  const uint32_t tileOff = (uint32_t)(uintptr_t)&tile[0][0];  // LDS byte offset
  const uint64_t gbase = (uint64_t)(uintptr_t)imgBase;

  // Issue one chunk's async copy: 3 instructions per wave, each moving
  // 512 contiguous bytes per wave (LDS layout mirrors global layout).
  auto issue_chunk = [&](int buf, int chunk) {
    const uint32_t gchunk = (uint32_t)chunk * CHUNK_BYTES;
    const uint32_t lbase = tileOff + (uint32_t)buf * CHUNK_BYTES;
    const uint32_t t16 = (uint32_t)tid * 16u;
#pragma unroll
    for (int pl = 0; pl < 3; ++pl) {
      const uint32_t off = t16 + (uint32_t)pl * (THREADS * 16);
      async_lds_b128(lbase + off, gbase, gchunk + off);
    }
  };

  // Kick off chunk 0 before building the LUTs so DMA overlaps setup.
  issue_chunk(0, chunk0);

  // Build separable stripe LUTs (removes per-pixel div/mod by runtime gb)
  for (int i = tid; i < TS; i += THREADS) {
    int d1 = i - st1;
    rowLUT[i] = (d1 >= 0 && d1 < ngb && (d1 % gb) < length) ? 1.0f : 0.0f;
    int d2 = i - st2;
    colLUT[i] = (d2 >= 0 && d2 < ngb && (d2 % gb) < length) ? 1.0f : 0.0f;
  }
  __syncthreads();  // LUTs visible to all waves (waves read tile bytes async-copied by other waves)

  for (int ci = 0; ci < CHUNKS; ++ci) {
    const int chunk = chunk0 + ci;
    const int cur = ci & 1;

    // Software pipeline: issue next chunk, then retire current chunk's
    // 3 async ops (ASYNCcnt completes in order -> wait <= 3 leaves only the
    // just-issued ones outstanding).
    if (ci + 1 < CHUNKS) {
      issue_chunk(cur ^ 1, chunk + 1);
      asm volatile("s_wait_asynccnt 0x3" ::: "memory");
    } else {
      asm volatile("s_wait_asynccnt 0x0" ::: "memory");
    }

    // ---- mask for this thread's 4 consecutive pixels ----
    const int p0 = chunk * PIX_CHUNK + tid * 4;  // pixel index within image
    const int y = p0 >> 9;                       // W == 512
    const int x = p0 & (WW - 1);
    const float yf = (float)y - 255.5f;          // (y + off_h) - (S-1)/2
    const float xf0 = (float)x - 255.5f;
    float m[4];
#pragma unroll
    for (int k = 0; k < 4; ++k) {
      const float xf = xf0 + (float)k;
      const float sx = cv * xf + sv * yf + 511.5f;
      const float sy = cv * yf - sv * xf + 511.5f;
      const float fsx = floorf(sx), fsy = floorf(sy);
      const float fx = sx - fsx, fy = sy - fsy;
      const int ix = (int)fsx, iy = (int)fsy;
      const int x0r = reflect2048(ix), x1r = reflect2048(ix + 1);
      const int y0r = reflect2048(iy), y1r = reflect2048(iy + 1);
      const float r0 = rowLUT[y0r], r1 = rowLUT[y1r];
      const float c0 = colLUT[x0r], c1 = colLUT[x1r];
      const float v00 = fmaxf(r0, c0), v01 = fmaxf(r0, c1);
      const float v10 = fmaxf(r1, c0), v11 = fmaxf(r1, c1);
      const float top = v00 + fx * (v01 - v00);
      const float bot = v10 + fx * (v11 - v10);
      m[k] = top + fy * (bot - top);
    }

    // ---- 12 floats from LDS, scale, 12 floats out (3x b128 NT store) ----
    const f4* src = (const f4*)&tile[cur][tid * 12];
    f4 a = src[0];
    f4 bv = src[1];
    f4 c = src[2];
    a.x *= m[0]; a.y *= m[0]; a.z *= m[0]; a.w *= m[1];
    bv.x *= m[1]; bv.y *= m[1]; bv.z *= m[2]; bv.w *= m[2];
    c.x *= m[2]; c.y *= m[3]; c.z *= m[3]; c.w *= m[3];

    f4* dst = (f4*)(outBase + (long long)p0 * CC);
    __builtin_nontemporal_store(a, dst + 0);
    __builtin_nontemporal_store(bv, dst + 1);
    __builtin_nontemporal_store(c, dst + 2);
  }
}

extern "C" void kernel_launch(void* const* d_in, const int* in_sizes, int n_in,
                              void* d_out, int out_size, void* d_ws, size_t ws_size,
                              hipStream_t stream) {
  const float* images = (const float*)d_in[0];
  const float* angles = (const float*)d_in[1];
  const int* gbp = (const int*)d_in[2];
  const int* s1 = (const int*)d_in[3];
  const int* s2 = (const int*)d_in[4];
  float* out = (float*)d_out;

  dim3 grid(BLOCKS_X, BB, 1);
  dim3 block(THREADS, 1, 1);
  hipLaunchKernelGGL(gridmask_fused_kernel, grid, block, 0, stream,
                     images, angles, gbp, s1, s2, out);
}